// logSigmoidApprox_23759759082177
// MI455X (gfx1250) — compile-verified
//
#include <hip/hip_runtime.h>

// Piecewise-linear log-sigmoid approximation, MI455X (gfx1250, wave32).
// Memory-bound streaming kernel: 4B in + 4B out per element @ 23.3 TB/s HBM
// => ~92us floor for 64*2048*2048 fp32. Strategy:
//  * x breakpoints are uniform (linspace) -> O(1) bucket index via
//    idx = clamp(floor((v - x0) * inv_h), 0, 63) instead of searchsorted.
//  * Precompute per-segment {x_lo, y_lo, slope} into LDS once per block:
//    hot loop does a single ds_load_b128 per element (broadcast on shared
//    buckets), leaving the VMEM pipe free for the stream.
//  * Native clang ext_vector float4 (b128) non-temporal loads/stores: data is
//    touched exactly once and exceeds the 192MB L2, so TH=NT keeps the stream
//    from polluting caches.
//  * Grid-stride loop, 256 threads (8 wave32s) per block; capped grid so each
//    thread streams many independent b128 transactions (deep LOADcnt queue).

#define NUM_BP 65
#define NUM_SEG 64

typedef __attribute__((ext_vector_type(4))) float v4f;

__device__ __forceinline__ float pwl_eval(float v,
                                          const v4f* __restrict__ seg,
                                          float x0, float xlast, float inv_h) {
    // Bucket index: trunc == floor for t >= 0; t < 0 clamps to 0 either way.
    float t = (v - x0) * inv_h;
    int idx = (int)t;
    idx = idx < 0 ? 0 : (idx > (NUM_SEG - 1) ? (NUM_SEG - 1) : idx);
    v4f s = seg[idx];               // ds_load_b128 {x_lo, y_lo, slope, pad}
    float interp = fmaf(v - s.x, s.z, s.y);
    // vals < x[0] -> identity; vals >= x[-1] -> 0; else interp.
    float r = (v < x0) ? v : ((v >= xlast) ? 0.0f : interp);
    return r;
}

__global__ __launch_bounds__(256) void logsig_pwl_kernel(
    const float* __restrict__ vals,
    const float* __restrict__ xb,
    const float* __restrict__ yb,
    float* __restrict__ out,
    long long n) {
    __shared__ v4f seg[NUM_SEG];             // 1 KB: {x_lo, y_lo, slope, 0}
    __shared__ float s_consts[3];            // x0, xlast, inv_h

    const int tid = threadIdx.x;
    if (tid < NUM_SEG) {
        float xl = xb[tid];
        float xh = xb[tid + 1];
        float yl = yb[tid];
        float yh = yb[tid + 1];
        float slope = (yh - yl) / (xh - xl);
        v4f e;
        e.x = xl; e.y = yl; e.z = slope; e.w = 0.0f;
        seg[tid] = e;
    }
    if (tid == 0) {
        float x0 = xb[0];
        float xl = xb[NUM_BP - 1];
        s_consts[0] = x0;
        s_consts[1] = xl;
        s_consts[2] = (float)NUM_SEG / (xl - x0);  // 1/h for uniform grid
    }
    __syncthreads();

    const float x0    = s_consts[0];
    const float xlast = s_consts[1];
    const float inv_h = s_consts[2];

    const long long gid     = (long long)blockIdx.x * blockDim.x + tid;
    const long long nthread = (long long)gridDim.x * blockDim.x;

    // Vectorized main body: one float4 per iteration, non-temporal b128.
    const long long n4 = n >> 2;
    const v4f* __restrict__ vin  = (const v4f*)vals;
    v4f* __restrict__       vout = (v4f*)out;

    #pragma unroll 2
    for (long long i = gid; i < n4; i += nthread) {
        v4f v = __builtin_nontemporal_load(&vin[i]);
        v4f r;
        r.x = pwl_eval(v.x, seg, x0, xlast, inv_h);
        r.y = pwl_eval(v.y, seg, x0, xlast, inv_h);
        r.z = pwl_eval(v.z, seg, x0, xlast, inv_h);
        r.w = pwl_eval(v.w, seg, x0, xlast, inv_h);
        __builtin_nontemporal_store(r, &vout[i]);
    }

    // Scalar tail (n % 4 elements; 0 for the reference shape, kept for safety).
    for (long long i = (n4 << 2) + gid; i < n; i += nthread) {
        float v = vals[i];
        out[i] = pwl_eval(v, seg, x0, xlast, inv_h);
    }
}

extern "C" void kernel_launch(void* const* d_in, const int* in_sizes, int n_in,
                              void* d_out, int out_size, void* d_ws, size_t ws_size,
                              hipStream_t stream) {
    (void)n_in; (void)d_ws; (void)ws_size; (void)out_size;

    const float* vals = (const float*)d_in[0];
    const float* xb   = (const float*)d_in[1];
    const float* yb   = (const float*)d_in[2];
    float* out        = (float*)d_out;

    const long long n  = (long long)in_sizes[0];   // 64*2048*2048 = 268,435,456
    const long long n4 = n >> 2;

    const int threads = 256;                       // 8 wave32s per block
    long long want_blocks = (n4 + threads - 1) / threads;
    if (want_blocks < 1) want_blocks = 1;
    // Cap grid: grid-stride so each thread streams ~16 independent float4s.
    const long long cap = 16384;
    int blocks = (int)(want_blocks < cap ? want_blocks : cap);

    logsig_pwl_kernel<<<blocks, threads, 0, stream>>>(vals, xb, yb, out, n);
}